// GCNNet_42855183679887
// MI455X (gfx1250) — compile-verified
//
#include <hip/hip_runtime.h>

#define N_NODES 100000
#define N_EDGES 1600000
#define DIM 128
#define N_LAYERS 4

// packed weights: [64 k-pairs][256] (pair-interleaved: Wp[p][2n+j] = W[2p+j][n])
// TDM pads 32 DWORDs after every 256 -> LDS pair-row stride 288 (conflict-free)
#define PAIR_STRIDE_LDS 288
#define WCHUNK_PAIRS  32                          // 64 k-rows per chunk
#define WCHUNK_FLOATS (WCHUNK_PAIRS * PAIR_STRIDE_LDS)
#define WCHUNK_BYTES  (WCHUNK_FLOATS * 4)
#define WPACK_FLOATS  (64 * 256)                  // one packed matrix (dense)

typedef __attribute__((ext_vector_type(2))) float v2f;
typedef __attribute__((ext_vector_type(8))) float v8f;
typedef __attribute__((ext_vector_type(4))) unsigned int v4u;
typedef __attribute__((ext_vector_type(8))) int v8i;
typedef __attribute__((ext_vector_type(4))) int v4i;

// ---------------------------------------------------------------- utilities
__global__ void zero_f32(float* __restrict__ p, int n) {
    int i = blockIdx.x * blockDim.x + threadIdx.x;
    if (i < n) p[i] = 0.0f;
}

__global__ void zero_f32_v4(float4* __restrict__ p, int n4) {
    int i = blockIdx.x * blockDim.x + threadIdx.x;
    if (i < n4) p[i] = make_float4(0.f, 0.f, 0.f, 0.f);
}

__global__ void count_deg(const int* __restrict__ src, const int* __restrict__ dst,
                          float* __restrict__ deg_out, float* __restrict__ deg_in, int nE) {
    int e = blockIdx.x * blockDim.x + threadIdx.x;
    if (e < nE) {
        unsafeAtomicAdd(&deg_out[src[e]], 1.0f);
        unsafeAtomicAdd(&deg_in[dst[e]], 1.0f);
    }
}

__global__ void finalize_norm(float* __restrict__ d, int n) {
    int i = blockIdx.x * blockDim.x + threadIdx.x;
    if (i < n) d[i] = rsqrtf(fmaxf(d[i], 1.0f));
}

// pair-interleave one 128x128 weight matrix: Wp[p*256 + 2n + j] = W[(2p+j)*128 + n]
__global__ void pack_w(const float* __restrict__ Wsrc, float* __restrict__ Wdst) {
    int i = blockIdx.x * blockDim.x + threadIdx.x;   // 64*128 threads
    int n = i & 127, p = i >> 7;
    float2 v = make_float2(Wsrc[(2 * p) * DIM + n], Wsrc[(2 * p + 1) * DIM + n]);
    *(float2*)&Wdst[p * 256 + 2 * n] = v;
}

// -------------------------------------------------- edge gather/scatter-add
// one wave per edge: 32 lanes x float4 = 512B row; fp32 atomics land in L2
__global__ void gather_scatter(const float4* __restrict__ h,
                               const float* __restrict__ norm_s,
                               const int* __restrict__ src,
                               const int* __restrict__ dst,
                               float* __restrict__ agg, int nE) {
    int gid  = blockIdx.x * blockDim.x + threadIdx.x;
    int e    = gid >> 5;
    int lane = gid & 31;
    if (e >= nE) return;
    int s = src[e], d = dst[e];
    float ns  = norm_s[s];
    float4 v  = h[(size_t)s * 32 + lane];
    float* ap = agg + (size_t)d * DIM + lane * 4;
    unsafeAtomicAdd(ap + 0, v.x * ns);
    unsafeAtomicAdd(ap + 1, v.y * ns);
    unsafeAtomicAdd(ap + 2, v.z * ns);
    unsafeAtomicAdd(ap + 3, v.w * ns);
}

// --------------------------------- TDM: stage 32 packed k-pair rows into LDS
// 2D tile 256x32 of 4-byte elements, pad 32 DWORDs after every 256 DWORDs
// -> LDS layout [32][288] (single-ds_load_b64 B fragments, conflict-free).
__device__ __forceinline__ void tdm_load_w(unsigned int lds_addr, const float* gptr) {
    unsigned long long ga = (unsigned long long)(uintptr_t)gptr;
    v4u g0;
    g0.x = 1u;                                          // count=1, user D#
    g0.y = lds_addr;                                    // LDS byte address
    g0.z = (unsigned int)(ga & 0xFFFFFFFFu);            // global_addr[31:0]
    g0.w = (unsigned int)((ga >> 32) & 0x01FFFFFFu)     // global_addr[56:32]
           | (2u << 30);                                // type = 2 (image)
    v8i g1;
    g1[0] = (int)((2u << 16)         // data_size = 4 bytes
                  | (1u << 20)       // pad_enable
                  | (7u << 22)       // pad_interval: 256 DWORDs
                  | (31u << 25));    // pad_amount:   32 DWORDs
    g1[1] = (int)(256u << 16);       // tensor_dim0 = 256 (bits 79:48)
    g1[2] = (int)(32u << 16);        // tensor_dim1 = 32  (bits 111:80)
    g1[3] = (int)(256u << 16);       // tile_dim0   = 256 (bits 127:112)
    g1[4] = 32;                      // tile_dim1   = 32  (bits 143:128)
    g1[5] = 256;                     // tensor_dim0_stride = 256 (bits 207:160)
    g1[6] = 0;
    g1[7] = 0;
    v4i z4 = {0, 0, 0, 0};
    v8i z8 = {0, 0, 0, 0, 0, 0, 0, 0};
    __builtin_amdgcn_tensor_load_to_lds(g0, g1, z4, z4, z8, 0);
}

// -------------------------------------------------------------- WMMA GEMM
// out[row,:] = (optional residual) + act( rowscale * (X[row,:] @ W) + bias )
// block = 128 threads = 4 waves; wave -> 16 rows x 128 cols (8 tiles of 16x16)
// K swept in steps of 4 via V_WMMA_F32_16X16X4_F32 (full fp32 precision).
// W (pair-packed) staged by the Tensor Data Mover, double-buffered.
__global__ __launch_bounds__(128)
void gemm_wmma(const float* __restrict__ X,       // [nRows,128]
               const float* __restrict__ Wp,      // packed [64][256]
               const float* __restrict__ bias,    // [128]
               const float* __restrict__ rownorm, // nullptr or [nRows]
               const float* __restrict__ hres,    // nullptr or [nRows,128]
               float* __restrict__ out,           // [nRows,128]
               int nRows, int doRelu) {
    extern __shared__ float Wsh[];                // 2 x [32][288] floats
    const unsigned int ldsBase = __builtin_amdgcn_groupstaticsize();

    const int wave   = threadIdx.x >> 5;
    const int lane   = threadIdx.x & 31;
    const int lhalf  = lane >> 4;       // 0: lanes 0-15, 1: lanes 16-31
    const int lmod   = lane & 15;
    const int rowBase = (blockIdx.x * 4 + wave) * 16;

    // A-fragment row, clamped: OOB rows compute garbage that is never stored
    const int arow  = rowBase + lmod;
    const int arowc = arow < nRows ? arow : (nRows - 1);
    const float* xrow = X + (size_t)arowc * DIM + 2 * lhalf;

    v8f c[8];
#pragma unroll
    for (int t = 0; t < 8; ++t) c[t] = (v8f){0.f,0.f,0.f,0.f,0.f,0.f,0.f,0.f};

    // prologue: DMA chunk 0 (k-rows 0..63), wait, then kick off chunk 1
    if (wave == 0) {
        tdm_load_w(ldsBase, Wp);
        __builtin_amdgcn_s_wait_tensorcnt((short)0);
    }
    __syncthreads();
    if (wave == 0) tdm_load_w(ldsBase + WCHUNK_BYTES, Wp + WCHUNK_PAIRS * 256);

    for (int half = 0; half < 2; ++half) {
        const float* Wc = Wsh + half * WCHUNK_FLOATS;
        for (int kk = 0; kk < 64; kk += 4) {
            v2f a = *(const v2f*)(xrow + half * 64 + kk);   // unconditional b64
            const int p = (kk >> 1) + lhalf;                // k-pair for this lane
#pragma unroll
            for (int t = 0; t < 8; ++t) {
                const int col = t * 16 + lmod;
                v2f b = *(const v2f*)(Wc + p * PAIR_STRIDE_LDS + 2 * col); // 1x b64
                c[t] = __builtin_amdgcn_wmma_f32_16x16x4_f32(
                           false, a, false, b, (short)0, c[t], false, false);
            }
        }
        if (half == 0) {                         // chunk-1 DMA completion fence
            if (wave == 0) __builtin_amdgcn_s_wait_tensorcnt((short)0);
            __syncthreads();
        }
    }

    // epilogue: rowscale, bias (+ReLU) (+residual)
    // C/D layout: row = rowBase + v + 8*lhalf, col = t*16 + lmod
    float rs[8];
#pragma unroll
    for (int v = 0; v < 8; ++v) {
        int row = rowBase + v + 8 * lhalf;
        int rc  = row < nRows ? row : (nRows - 1);
        rs[v] = rownorm ? rownorm[rc] : 1.0f;
    }
#pragma unroll
    for (int t = 0; t < 8; ++t) {
        const int col = t * 16 + lmod;
        const float bv = bias[col];
#pragma unroll
        for (int v = 0; v < 8; ++v) {
            const int row = rowBase + v + 8 * lhalf;
            if (row < nRows) {
                float z = fmaf(c[t][v], rs[v], bv);
                if (doRelu) z = fmaxf(z, 0.0f);
                if (hres)   z += hres[(size_t)row * DIM + col];
                out[(size_t)row * DIM + col] = z;
            }
        }
    }
}

// ------------------------------------------------------------------ driver
extern "C" void kernel_launch(void* const* d_in, const int* in_sizes, int n_in,
                              void* d_out, int out_size, void* d_ws, size_t ws_size,
                              hipStream_t stream) {
    const float* h_in    = (const float*)d_in[0];
    const int*   src     = (const int*)  d_in[1];
    const int*   dst     = (const int*)  d_in[2];
    const float* W_embed = (const float*)d_in[3];
    const float* b_embed = (const float*)d_in[4];
    const float* Ws      = (const float*)d_in[5];
    const float* bs      = (const float*)d_in[6];
    float* out = (float*)d_out;

    float* ws      = (float*)d_ws;
    float* norm_s  = ws;                                  // [N]
    float* norm_d  = ws + N_NODES;                        // [N]
    float* agg     = ws + 2 * N_NODES;                    // [N,128]
    float* wpacked = agg + (size_t)N_NODES * DIM;         // 5 x [64][256]

    // degrees -> norms (float atomics, L2-resident)
    zero_f32<<<(2 * N_NODES + 255) / 256, 256, 0, stream>>>(norm_s, 2 * N_NODES);
    count_deg<<<(N_EDGES + 255) / 256, 256, 0, stream>>>(src, dst, norm_s, norm_d, N_EDGES);
    finalize_norm<<<(2 * N_NODES + 255) / 256, 256, 0, stream>>>(norm_s, 2 * N_NODES);

    // pair-interleave the 5 weight matrices (one-time, 64KB each)
    pack_w<<<32, 256, 0, stream>>>(W_embed, wpacked);
    for (int L = 0; L < N_LAYERS; ++L)
        pack_w<<<32, 256, 0, stream>>>(Ws + (size_t)L * DIM * DIM,
                                       wpacked + (size_t)(L + 1) * WPACK_FLOATS);

    const dim3 gGemm((N_NODES + 63) / 64);
    const size_t ldsBytes = 2 * WCHUNK_BYTES;

    // h = h_in @ W_embed + b_embed   (into d_out; updated in place per layer)
    gemm_wmma<<<gGemm, 128, ldsBytes, stream>>>(h_in, wpacked, b_embed,
                                                nullptr, nullptr, out, N_NODES, 0);

    const int n4 = N_NODES * DIM / 4;
    for (int L = 0; L < N_LAYERS; ++L) {
        zero_f32_v4<<<(n4 + 255) / 256, 256, 0, stream>>>((float4*)agg, n4);
        gather_scatter<<<(N_EDGES * 32 + 255) / 256, 256, 0, stream>>>(
            (const float4*)out, norm_s, src, dst, agg, N_EDGES);
        // h = h + relu( (agg * norm_d) @ Ws[L] + bs[L] )   -- in place
        gemm_wmma<<<gGemm, 128, ldsBytes, stream>>>(agg,
                                                    wpacked + (size_t)(L + 1) * WPACK_FLOATS,
                                                    bs + (size_t)L * DIM,
                                                    norm_d, out, out, N_NODES, 1);
    }
}